// NormPool2d_7267084665311
// MI455X (gfx1250) — compile-verified
//
#include <hip/hip_runtime.h>
#include <cstdint>
#include <cstddef>

// NormPool2d (3x3, reflect pad=1): out = mean - unbiased_std over each window.
// Memory-bound: ~616MB traffic @ 23.3TB/s => ~27us floor. Strategy: async
// global->LDS staging of a 34x34 halo tile (gfx1250 ASYNCcnt path), VALU
// sliding-window reduction, coalesced stores.

#define IMG_H 224
#define IMG_W 224
#define TILE 32
#define HALO (TILE + 2)          // 34
#define HALO_ELEMS (HALO * HALO) // 1156
#define NTHREADS 256

#define AS1 __attribute__((address_space(1)))
#define AS3 __attribute__((address_space(3)))

__device__ __forceinline__ int reflect1(int i, int n) {
    // reflect padding with pad=1: -1 -> 1, n -> n-2
    if (i < 0) i = -i;
    if (i >= n) i = 2 * n - 2 - i;
    return i;
}

__global__ __launch_bounds__(NTHREADS)
void normpool2d_kernel(const float* __restrict__ x, float* __restrict__ out) {
    __shared__ float tile[HALO_ELEMS];

    const int tx0   = blockIdx.x * TILE;
    const int ty0   = blockIdx.y * TILE;
    const int plane = blockIdx.z;
    const size_t pbase = (size_t)plane * (size_t)(IMG_H * IMG_W);
    const float* __restrict__ xp = x + pbase;

    const int t = threadIdx.x;

    // ---- Stage halo tile into LDS (async-to-LDS, per-lane reflect addressing) ----
    for (int idx = t; idx < HALO_ELEMS; idx += NTHREADS) {
        const int hy = idx / HALO;
        const int hx = idx - hy * HALO;
        const int gy = reflect1(ty0 + hy - 1, IMG_H);
        const int gx = reflect1(tx0 + hx - 1, IMG_W);
        const float* gsrc = xp + (size_t)gy * IMG_W + gx;
#if __has_builtin(__builtin_amdgcn_global_load_async_to_lds_b32)
        __builtin_amdgcn_global_load_async_to_lds_b32(
            (AS1 int*)const_cast<float*>(gsrc),
            (AS3 int*)&tile[idx],
            /*imm offset*/ 0, /*cpol*/ 0);
#else
        tile[idx] = *gsrc;  // fallback: plain load + ds_store
#endif
    }
#if __has_builtin(__builtin_amdgcn_s_wait_asynccnt)
    __builtin_amdgcn_s_wait_asynccnt(0);
#else
    asm volatile("s_wait_asynccnt 0" ::: "memory");
#endif
    __syncthreads();

    // ---- Compute: each thread does 4 vertically adjacent outputs ----
    const int c     = t & (TILE - 1);     // output column within tile (lane-consecutive)
    const int rbase = (t >> 5) * 4;       // output row base (8 groups * 4 rows = 32)

    float rs[6], rq[6];
#pragma unroll
    for (int i = 0; i < 6; ++i) {
        const float* rp = &tile[(rbase + i) * HALO + c];
        const float a = rp[0];
        const float b = rp[1];
        const float d = rp[2];
        rs[i] = a + b + d;
        rq[i] = fmaf(a, a, fmaf(b, b, d * d));
    }

    float* __restrict__ op = out + pbase;
#pragma unroll
    for (int j = 0; j < 4; ++j) {
        const float s = rs[j] + rs[j + 1] + rs[j + 2];
        const float q = rq[j] + rq[j + 1] + rq[j + 2];
        const float mean = s * (1.0f / 9.0f);
        float var = (q - 9.0f * mean * mean) * (1.0f / 8.0f);
        var = fmaxf(var, 0.0f);
        const float res = mean - sqrtf(var);
        op[(size_t)(ty0 + rbase + j) * IMG_W + (tx0 + c)] = res;
    }
}

extern "C" void kernel_launch(void* const* d_in, const int* in_sizes, int n_in,
                              void* d_out, int out_size, void* d_ws, size_t ws_size,
                              hipStream_t stream) {
    const float* x = (const float*)d_in[0];
    float* out = (float*)d_out;

    const int planes = in_sizes[0] / (IMG_H * IMG_W);  // 16*96 = 1536
    dim3 grid(IMG_W / TILE, IMG_H / TILE, planes);     // (7, 7, 1536)
    normpool2d_kernel<<<grid, NTHREADS, 0, stream>>>(x, out);
}